// TransformerConvolution_2267742732893
// MI455X (gfx1250) — compile-verified
//
#include <hip/hip_runtime.h>
#include <hip/hip_bf16.h>

// Problem constants (reference): B=2, N=512, IN=64, OUT=128, H=8, D=16
#define BB   2
#define NN   512
#define CIN  64
#define DOUT 128
#define NH   8
#define HD   16

typedef float v2f __attribute__((ext_vector_type(2)));
typedef float v8f __attribute__((ext_vector_type(8)));

// FP32 matrix core op: D(16x16) = A(16x4) * B(4x16) + C
__device__ __forceinline__ v8f wmma4(v2f a, v2f b, v8f c) {
    return __builtin_amdgcn_wmma_f32_16x16x4_f32(
        /*neg_a=*/false, a, /*neg_b=*/false, b,
        /*c_mod=*/(short)0, c, /*reuse_a=*/false, /*reuse_b=*/false);
}

// -------------------------------------------------------------------------
// Kernel 1: q,k,v = concat(node,hidden) @ W_{q,k,v}   (B*N x 128, K=128)
//           t[b][n][h][c] = sum_d q[b,n,h*16+d] * We_k[c, h*16+d]
// grid = B*N/16 = 64 blocks, 128 threads (4 waves)
// -------------------------------------------------------------------------
__global__ void __launch_bounds__(128)
prep_qkvt(const float* __restrict__ node, const float* __restrict__ hidden,
          const float* __restrict__ Wq, const float* __restrict__ Wk,
          const float* __restrict__ Wv, const float* __restrict__ Wek,
          float* __restrict__ q, float* __restrict__ k, float* __restrict__ v,
          float* __restrict__ t)
{
    __shared__ float z_s[16 * 132];   // 16 rows x 128 cols, pitch 132 (bank-safe)
    __shared__ float q_s[16 * 132];

    const int tid  = threadIdx.x;
    const int wave = tid >> 5;
    const int lane = tid & 31;
    const int l15  = lane & 15;
    const int hi   = lane >> 4;       // 0: K=0,1 / rows 0-7 ; 1: K=2,3 / rows 8-15
    const int b    = blockIdx.x >> 5;
    const int n0   = (blockIdx.x & 31) * 16;

    // stage z = [node | hidden]
    for (int idx = tid; idx < 16 * 32; idx += 128) {
        int row = idx >> 5, c4 = idx & 31;
        const float* src = (c4 < 16)
            ? node   + ((size_t)(b * NN + n0 + row)) * CIN + c4 * 4
            : hidden + ((size_t)(b * NN + n0 + row)) * CIN + (c4 - 16) * 4;
        *(float4*)&z_s[row * 132 + c4 * 4] = *(const float4*)src;
    }
    __syncthreads();

    if (wave < 3) {
        const float* W   = (wave == 0) ? Wq : (wave == 1) ? Wk : Wv;
        float*       dst = (wave == 0) ? q  : (wave == 1) ? k  : v;
        for (int ct = 0; ct < 8; ++ct) {               // 8 column tiles of 16
            v8f acc = {};
            for (int k0 = 0; k0 < 128; k0 += 4) {
                int kk = k0 + 2 * hi;
                v2f a = *(const v2f*)&z_s[l15 * 132 + kk];
                v2f bb; bb.x = W[kk * DOUT + ct * 16 + l15];
                        bb.y = W[(kk + 1) * DOUT + ct * 16 + l15];
                acc = wmma4(a, bb, acc);
            }
            for (int r = 0; r < 8; ++r) {
                int row = hi ? r + 8 : r;
                float val = acc[r];
                dst[((size_t)(b * NN + n0 + row)) * DOUT + ct * 16 + l15] = val;
                if (wave == 0) q_s[row * 132 + ct * 16 + l15] = val;
            }
        }
    }
    __syncthreads();

    // t phase: per head h, (16n x 16d) @ (16d x 64c) with B[d][c] = We_k[c][h*16+d]
    for (int hh = 0; hh < 2; ++hh) {
        int h = wave + hh * 4;
        for (int ct = 0; ct < 4; ++ct) {
            v8f acc = {};
            for (int k0 = 0; k0 < 16; k0 += 4) {
                int kk = k0 + 2 * hi;
                v2f a = *(const v2f*)&q_s[l15 * 132 + h * HD + kk];
                v2f bb; bb.x = Wek[(ct * 16 + l15) * DOUT + h * HD + kk];
                        bb.y = Wek[(ct * 16 + l15) * DOUT + h * HD + kk + 1];
                acc = wmma4(a, bb, acc);
            }
            for (int r = 0; r < 8; ++r) {
                int row = hi ? r + 8 : r;
                // coalesced: consecutive c per lane
                t[(((size_t)(b * NN + n0 + row)) * NH + h) * CIN + ct * 16 + l15] = acc[r];
            }
        }
    }
}

// -------------------------------------------------------------------------
// Kernel 2: per 16x16 (i,j) tile, all 8 heads:
//   s2[j][i][h] = edge[b,i,j,:] (16x64) @ t[b,j,:,:] (64x8)   -- WMMA per j
//   qk[h][i][j] = q_h (16x16) @ k_h^T (16x16)                 -- WMMA per h
//   e = adj * exp(0.25*(qk+s2)); rowsum += partial sums (atomic)
// grid = B*32*32 = 2048 blocks, 128 threads
// -------------------------------------------------------------------------
__global__ void __launch_bounds__(128)
scores_kernel(const float* __restrict__ edge, const float* __restrict__ adj,
              const float* __restrict__ q, const float* __restrict__ k,
              const float* __restrict__ t, float* __restrict__ e,
              float* __restrict__ rowsum)
{
    __shared__ float E_s[16 * 516];    // [i][jloc(0..7)*64 + c], pitch 516 (33KB)
    __shared__ float s2_s[16 * 132];   // [j][i*8 + h], pitch 132 (8.4KB)

    const int tid  = threadIdx.x;
    const int wave = tid >> 5;
    const int lane = tid & 31;
    const int l15  = lane & 15;
    const int hi   = lane >> 4;
    const int id   = blockIdx.x;
    const int jt = id & 31, it = (id >> 5) & 31, b = id >> 10;
    const int i0 = it * 16, j0 = jt * 16;

    for (int jh = 0; jh < 2; ++jh) {
        // stage 16 i-rows x (8 j x 64 c) contiguous slabs
        for (int idx = tid; idx < 16 * 128; idx += 128) {
            int row = idx >> 7, c4 = idx & 127;
            const float* src =
                edge + (((size_t)(b * NN + i0 + row)) * NN + j0 + jh * 8) * CIN + c4 * 4;
            *(float4*)&E_s[row * 516 + c4 * 4] = *(const float4*)src;
            // overlap next slab's HBM fetch with this slab's WMMA work
            if (jh == 0 && (c4 & 7) == 0)
                __builtin_prefetch(src + 8 * CIN, 0, 0);   // -> global_prefetch_b8
        }
        __syncthreads();

        for (int jj = 0; jj < 2; ++jj) {
            int jloc = wave + jj * 4;
            int j = j0 + jh * 8 + jloc;
            v8f acc = {};
            for (int k0 = 0; k0 < 64; k0 += 4) {
                int kk = k0 + 2 * hi;
                v2f a = *(const v2f*)&E_s[l15 * 516 + jloc * 64 + kk];
                // B[c][h] = t[b][j][h][c]; cols 8..15 mirror 0..7 (ignored)
                v2f bb = *(const v2f*)&t[(((size_t)(b * NN + j)) * NH + (l15 & 7)) * CIN + kk];
                acc = wmma4(a, bb, acc);
            }
            if (l15 < 8) {
                for (int r = 0; r < 8; ++r) {
                    int row = hi ? r + 8 : r;
                    s2_s[(jh * 8 + jloc) * 132 + row * 8 + l15] = acc[r];
                }
            }
        }
        __syncthreads();
    }

    // q.k^T per head, then combine + masked exp + rowsum atomics
    for (int hh = 0; hh < 2; ++hh) {
        int h = wave + hh * 4;
        v8f acc = {};
        for (int k0 = 0; k0 < HD; k0 += 4) {
            int kk = k0 + 2 * hi;
            v2f a  = *(const v2f*)&q[((size_t)(b * NN + i0 + l15)) * DOUT + h * HD + kk];
            v2f bb = *(const v2f*)&k[((size_t)(b * NN + j0 + l15)) * DOUT + h * HD + kk];
            acc = wmma4(a, bb, acc);
        }
        for (int r = 0; r < 8; ++r) {
            int row = hi ? r + 8 : r;
            int gi = i0 + row, gj = j0 + l15;
            float s  = (acc[r] + s2_s[l15 * 132 + row * 8 + h]) * 0.25f; // 1/sqrt(16)
            float av = adj[((size_t)(b * NN + gi)) * NN + gj];
            float ev = av * __expf(s);
            e[(((size_t)(b * NH + h)) * NN + gi) * NN + gj] = ev;
            float red = ev;
            red += __shfl_xor(red, 1, 16);
            red += __shfl_xor(red, 2, 16);
            red += __shfl_xor(red, 4, 16);
            red += __shfl_xor(red, 8, 16);
            if (l15 == 0)
                atomicAdd(&rowsum[((size_t)(b * NH + h)) * NN + gi], red);
        }
    }
}

// -------------------------------------------------------------------------
// Kernel 3: ctx[b,i,h*16+d] = (1/rowsum) * sum_j e[b,h,i,j] * v[b,j,h*16+d]
// 4 waves per block, one i-tile each; v_h^T staged once in LDS (conflict-free
// pitch 516, B-operand becomes ds_load_b64). grid = B*H*8 = 128 blocks.
// -------------------------------------------------------------------------
__global__ void __launch_bounds__(128)
attnv_kernel(const float* __restrict__ e, const float* __restrict__ v,
             const float* __restrict__ rowsum, float* __restrict__ ctx)
{
    __shared__ float v_s[16 * 516];   // [d][j], pitch 516 (33KB)
    const int tid  = threadIdx.x;
    const int wave = tid >> 5;
    const int lane = tid & 31;
    const int l15 = lane & 15, hi = lane >> 4;
    const int id = blockIdx.x;
    const int itq = id & 7, h = (id >> 3) & 7, b = id >> 6;
    const int i0 = (itq * 4 + wave) * 16;
    const size_t ebase = ((size_t)(b * NH + h)) * NN;

    // stage v[b, :, h*16 : h*16+16] transposed: v_s[d][j]
    for (int idx = tid; idx < NN * 4; idx += 128) {
        int j = idx >> 2, dq = idx & 3;
        float4 val = *(const float4*)&v[((size_t)(b * NN + j)) * DOUT + h * HD + dq * 4];
        v_s[(dq * 4 + 0) * 516 + j] = val.x;
        v_s[(dq * 4 + 1) * 516 + j] = val.y;
        v_s[(dq * 4 + 2) * 516 + j] = val.z;
        v_s[(dq * 4 + 3) * 516 + j] = val.w;
    }
    __syncthreads();

    v8f acc = {};
    for (int j0 = 0; j0 < NN; j0 += 4) {
        int kk = j0 + 2 * hi;
        v2f a  = *(const v2f*)&e[(ebase + i0 + l15) * NN + kk];
        v2f bb = *(const v2f*)&v_s[l15 * 516 + kk];
        acc = wmma4(a, bb, acc);
    }
    for (int r = 0; r < 8; ++r) {
        int row = hi ? r + 8 : r;
        float inv = 1.0f / rowsum[ebase + i0 + row];
        ctx[((size_t)(b * NN + i0 + row)) * DOUT + h * HD + l15] = acc[r] * inv;
    }
}

// -------------------------------------------------------------------------
// Kernel 4: out = relu(relu(ctx @ Wo @ W1 + b1) @ W2 + b2)
// grid = 64 blocks, 128 threads; three LDS-staged 16x128x128 WMMA GEMMs.
// -------------------------------------------------------------------------
__device__ __forceinline__ void
tile_gemm128(const float* in_s, const float* __restrict__ W,
             const float* __restrict__ bias, bool relu,
             float* out_s, float* out_g, size_t out_row0,
             int wave, int l15, int hi)
{
    for (int tt = 0; tt < 2; ++tt) {
        int ct = wave * 2 + tt;
        v8f acc = {};
        for (int k0 = 0; k0 < 128; k0 += 4) {
            int kk = k0 + 2 * hi;
            v2f a = *(const v2f*)&in_s[l15 * 132 + kk];
            v2f bb; bb.x = W[kk * DOUT + ct * 16 + l15];
                    bb.y = W[(kk + 1) * DOUT + ct * 16 + l15];
            acc = wmma4(a, bb, acc);
        }
        float bv = bias ? bias[ct * 16 + l15] : 0.0f;
        for (int r = 0; r < 8; ++r) {
            int row = hi ? r + 8 : r;
            float val = acc[r] + bv;
            if (relu) val = fmaxf(val, 0.0f);
            if (out_s) out_s[row * 132 + ct * 16 + l15] = val;
            else       out_g[(out_row0 + row) * DOUT + ct * 16 + l15] = val;
        }
    }
}

__global__ void __launch_bounds__(128)
ffn_kernel(const float* __restrict__ ctx, const float* __restrict__ Wo,
           const float* __restrict__ W1, const float* __restrict__ b1,
           const float* __restrict__ W2, const float* __restrict__ b2,
           float* __restrict__ out)
{
    __shared__ float bufA[16 * 132];
    __shared__ float bufB[16 * 132];
    const int tid = threadIdx.x, wave = tid >> 5, lane = tid & 31;
    const int l15 = lane & 15, hi = lane >> 4;
    const int b = blockIdx.x >> 5, n0 = (blockIdx.x & 31) * 16;

    for (int idx = tid; idx < 16 * 32; idx += 128) {
        int row = idx >> 5, c4 = idx & 31;
        *(float4*)&bufA[row * 132 + c4 * 4] =
            *(const float4*)&ctx[((size_t)(b * NN + n0 + row)) * DOUT + c4 * 4];
    }
    __syncthreads();
    tile_gemm128(bufA, Wo, nullptr, false, bufB, nullptr, 0, wave, l15, hi);
    __syncthreads();
    tile_gemm128(bufB, W1, b1, true, bufA, nullptr, 0, wave, l15, hi);
    __syncthreads();
    tile_gemm128(bufA, W2, b2, true, nullptr, out, (size_t)(b * NN + n0), wave, l15, hi);
}

// -------------------------------------------------------------------------
extern "C" void kernel_launch(void* const* d_in, const int* in_sizes, int n_in,
                              void* d_out, int out_size, void* d_ws, size_t ws_size,
                              hipStream_t stream)
{
    const float* node   = (const float*)d_in[0];
    const float* edge   = (const float*)d_in[1];
    /* d_in[2] graph_fts: unused by the reference math */
    const float* hidden = (const float*)d_in[3];
    const float* adj    = (const float*)d_in[4];
    const float* Wq  = (const float*)d_in[5];
    const float* Wk  = (const float*)d_in[6];
    const float* Wv  = (const float*)d_in[7];
    const float* Wo  = (const float*)d_in[8];
    const float* Wek = (const float*)d_in[9];
    const float* W1  = (const float*)d_in[10];
    const float* b1  = (const float*)d_in[11];
    const float* W2  = (const float*)d_in[12];
    const float* b2  = (const float*)d_in[13];
    float* out = (float*)d_out;

    float* ws = (float*)d_ws;
    size_t o = 0;
    float* q      = ws + o; o += (size_t)BB * NN * DOUT;          // 131072
    float* kbuf   = ws + o; o += (size_t)BB * NN * DOUT;          // 131072
    float* vbuf   = ws + o; o += (size_t)BB * NN * DOUT;          // 131072
    float* t      = ws + o; o += (size_t)BB * NN * NH * CIN + 64; // 524288 (+slack)
    float* e      = ws + o; o += (size_t)BB * NH * NN * NN;       // 4194304 (16.8MB, L2-resident)
    float* rowsum = ws + o; o += (size_t)BB * NH * NN;            // 8192
    float* ctx    = ws + o; o += (size_t)BB * NN * DOUT;          // 131072

    hipMemsetAsync(rowsum, 0, (size_t)BB * NH * NN * sizeof(float), stream);

    prep_qkvt    <<<BB * (NN / 16),              128, 0, stream>>>(node, hidden, Wq, Wk, Wv, Wek,
                                                                   q, kbuf, vbuf, t);
    scores_kernel<<<BB * (NN / 16) * (NN / 16),  128, 0, stream>>>(edge, adj, q, kbuf, t, e, rowsum);
    attnv_kernel <<<BB * NH * (NN / 64),         128, 0, stream>>>(e, vbuf, rowsum, ctx);
    ffn_kernel   <<<BB * (NN / 16),              128, 0, stream>>>(ctx, Wo, W1, b1, W2, b2, out);
}